// FractalNetwork_17489106829982
// MI455X (gfx1250) — compile-verified
//
#include <hip/hip_runtime.h>
#include <math.h>

// Problem constants (from reference): B=2048, N=128, D=64, H=2D=128, DEPTH=3
#define FB 2048
#define FN 128
#define FD 64
#define FH 128

typedef float v2f __attribute__((ext_vector_type(2)));
typedef float v8f __attribute__((ext_vector_type(8)));

// D(16x16,f32) = A(16x4,f32) * B(4x16,f32) + C  -- exact fp32 matrix op on CDNA5
__device__ __forceinline__ v8f wmma4(v2f a, v2f b, v8f c) {
    return __builtin_amdgcn_wmma_f32_16x16x4_f32(false, a, false, b, (short)0, c,
                                                 false, false);
}

__device__ __forceinline__ float gelu_exact(float x) {
    return 0.5f * x * (1.0f + erff(x * 0.70710678118654752f));
}

// ---------------------------------------------------------------------------
// Stage A: per-node MLP + shared encoder.
//   h = LN(x @ W1 + b1); h = gelu(h); p = h @ W2 + b2; m = p @ encW + encb
// One wave32 per (batch-tile of 16, node). Grid: (N, B/16).
// ---------------------------------------------------------------------------
__global__ __launch_bounds__(32) void fractal_stageA(
    const float* __restrict__ x,    // [B,N,D]
    const float* __restrict__ W1,   // [N,D,H]
    const float* __restrict__ b1,   // [N,H]
    const float* __restrict__ lng,  // [N,H]
    const float* __restrict__ lnb,  // [N,H]
    const float* __restrict__ W2,   // [N,H,D]
    const float* __restrict__ b2,   // [N,D]
    const float* __restrict__ encW, // [D,D]
    const float* __restrict__ encb, // [D]
    float* __restrict__ mout)       // [B,N,D]
{
    const int n    = blockIdx.x;
    const int bt   = blockIdx.y;
    const int lane = threadIdx.x;
    const int lrow = lane & 15;   // A-layout: lane holds row M = lrow
    const int half = lane >> 4;   // selects K pair {0,1} vs {2,3} of each K4 step

    __shared__ __align__(16) float hls[16 * 132];  // h tile, padded stride
    __shared__ __align__(16) float pls[16 * 68];   // p tile, padded stride

    // ---- A fragments of the x tile (16 rows x K=64): 16 K-steps of v2f ----
    v2f ax[16];
    {
        const float* xr = x + ((size_t)(bt * 16 + lrow) * FN + n) * FD;
        #pragma unroll
        for (int s = 0; s < 16; ++s) {
            const int k = 4 * s + 2 * half;
            ax[s][0] = xr[k];
            ax[s][1] = xr[k + 1];
        }
    }

    // ---- h = x @ W1 + b1 : 16 x 128 as 8 column tiles ----
    const float* W1n = W1 + (size_t)n * FD * FH;
    v8f h[8];
    #pragma unroll
    for (int ct = 0; ct < 8; ++ct) {
        const int col = ct * 16 + lrow;  // B/C layout: lane holds column
        v8f acc = {};
        #pragma unroll
        for (int s = 0; s < 16; ++s) {
            const int k = 4 * s + 2 * half;
            v2f bf;
            bf[0] = W1n[k * FH + col];
            bf[1] = W1n[(k + 1) * FH + col];
            acc = wmma4(ax[s], bf, acc);
        }
        const float bias = b1[n * FH + col];
        #pragma unroll
        for (int r = 0; r < 8; ++r) acc[r] += bias;
        h[ct] = acc;
    }

    // ---- LayerNorm over H per row (row M = r + 8*half lives in one 16-lane half) ----
    float s1[8], s2[8];
    #pragma unroll
    for (int r = 0; r < 8; ++r) { s1[r] = 0.f; s2[r] = 0.f; }
    #pragma unroll
    for (int ct = 0; ct < 8; ++ct) {
        #pragma unroll
        for (int r = 0; r < 8; ++r) {
            const float v = h[ct][r];
            s1[r] += v;
            s2[r] += v * v;
        }
    }
    #pragma unroll
    for (int r = 0; r < 8; ++r) {
        #pragma unroll
        for (int mk = 1; mk < 16; mk <<= 1) {
            s1[r] += __shfl_xor(s1[r], mk, 16);
            s2[r] += __shfl_xor(s2[r], mk, 16);
        }
    }
    float mu[8], rstd[8];
    #pragma unroll
    for (int r = 0; r < 8; ++r) {
        mu[r] = s1[r] * (1.0f / FH);
        const float var = s2[r] * (1.0f / FH) - mu[r] * mu[r];  // biased var
        rstd[r] = rsqrtf(var + 1e-5f);
    }
    // normalize + affine + exact GELU, spill to LDS in plain row-major
    #pragma unroll
    for (int ct = 0; ct < 8; ++ct) {
        const int col = ct * 16 + lrow;
        const float g  = lng[n * FH + col];
        const float bb = lnb[n * FH + col];
        #pragma unroll
        for (int r = 0; r < 8; ++r) {
            float v = (h[ct][r] - mu[r]) * rstd[r] * g + bb;
            hls[(r + 8 * half) * 132 + col] = gelu_exact(v);
        }
    }
    __syncthreads();

    // ---- p = h @ W2 + b2 : 16 x 64 (K=128) ----
    const float* W2n = W2 + (size_t)n * FH * FD;
    #pragma unroll
    for (int ct = 0; ct < 4; ++ct) {
        const int col = ct * 16 + lrow;
        v8f acc = {};
        #pragma unroll
        for (int s = 0; s < 32; ++s) {
            const int k = 4 * s + 2 * half;
            const v2f af = *(const v2f*)&hls[lrow * 132 + k];  // re-fragment A
            v2f bf;
            bf[0] = W2n[k * FD + col];
            bf[1] = W2n[(k + 1) * FD + col];
            acc = wmma4(af, bf, acc);
        }
        const float bias = b2[n * FD + col];
        #pragma unroll
        for (int r = 0; r < 8; ++r) pls[(r + 8 * half) * 68 + col] = acc[r] + bias;
    }
    __syncthreads();

    // ---- m = p @ encW + encb, streamed to global scratch ----
    float* mo = mout + ((size_t)bt * 16 * FN + n) * FD;
    #pragma unroll
    for (int ct = 0; ct < 4; ++ct) {
        const int col = ct * 16 + lrow;
        v8f acc = {};
        #pragma unroll
        for (int s = 0; s < 16; ++s) {
            const int k = 4 * s + 2 * half;
            const v2f af = *(const v2f*)&pls[lrow * 68 + k];
            v2f bf;
            bf[0] = encW[k * FD + col];
            bf[1] = encW[(k + 1) * FD + col];
            acc = wmma4(af, bf, acc);
        }
        const float bias = encb[col];
        #pragma unroll
        for (int r = 0; r < 8; ++r) {
            const int row = r + 8 * half;
            mo[(size_t)row * FN * FD + col] = acc[r] + bias;
        }
    }
}

// ---------------------------------------------------------------------------
// Stage B: cross-node aggregation + shared decoder.
//   r[b] = conn @ m[b]  (128x128 @ 128x64);  x[b] = r[b] @ decW + decb
// One wave32 per batch element. Grid: (B).
// ---------------------------------------------------------------------------
__global__ __launch_bounds__(32) void fractal_stageB(
    const float* __restrict__ m,    // [B,N,D]
    const float* __restrict__ conn, // [N,N] for this depth
    const float* __restrict__ decW, // [D,D]
    const float* __restrict__ decb, // [D]
    float* __restrict__ xout)       // [B,N,D]
{
    const int b    = blockIdx.x;
    const int lane = threadIdx.x;
    const int lrow = lane & 15;
    const int half = lane >> 4;

    __shared__ __align__(16) float rls[128 * 68];  // r[b] staged for decode

    const float* mb = m + (size_t)b * FN * FD;

    // phase 1: r = conn @ m[b]
    for (int it = 0; it < 8; ++it) {
        // A fragments of conn row-tile (16 x 128): shared across all b, L2-hot
        v2f a[32];
        #pragma unroll
        for (int s = 0; s < 32; ++s) {
            const int k = 4 * s + 2 * half;
            const float* cr = conn + (size_t)(it * 16 + lrow) * FN + k;
            a[s][0] = cr[0];
            a[s][1] = cr[1];
        }
        #pragma unroll
        for (int ct = 0; ct < 4; ++ct) {
            const int col = ct * 16 + lrow;
            v8f acc = {};
            #pragma unroll
            for (int s = 0; s < 32; ++s) {
                const int k = 4 * s + 2 * half;
                v2f bf;
                bf[0] = mb[(size_t)k * FD + col];
                bf[1] = mb[(size_t)(k + 1) * FD + col];
                acc = wmma4(a[s], bf, acc);
            }
            #pragma unroll
            for (int r = 0; r < 8; ++r)
                rls[(it * 16 + r + 8 * half) * 68 + col] = acc[r];
        }
    }
    __syncthreads();

    // phase 2: x = r @ decW + decb
    for (int it = 0; it < 8; ++it) {
        #pragma unroll
        for (int ct = 0; ct < 4; ++ct) {
            const int col = ct * 16 + lrow;
            v8f acc = {};
            #pragma unroll
            for (int s = 0; s < 16; ++s) {
                const int k = 4 * s + 2 * half;
                const v2f af = *(const v2f*)&rls[(it * 16 + lrow) * 68 + k];
                v2f bf;
                bf[0] = decW[k * FD + col];
                bf[1] = decW[(k + 1) * FD + col];
                acc = wmma4(af, bf, acc);
            }
            const float bias = decb[col];
            #pragma unroll
            for (int r = 0; r < 8; ++r) {
                const int row = it * 16 + r + 8 * half;
                xout[((size_t)b * FN + row) * FD + col] = acc[r] + bias;
            }
        }
    }
}

// ---------------------------------------------------------------------------
extern "C" void kernel_launch(void* const* d_in, const int* in_sizes, int n_in,
                              void* d_out, int out_size, void* d_ws, size_t ws_size,
                              hipStream_t stream) {
    (void)in_sizes; (void)n_in; (void)out_size; (void)ws_size;

    const float* x0   = (const float*)d_in[0];   // node_states [B,N,D]
    const float* W1   = (const float*)d_in[1];   // [N,D,H]
    const float* b1   = (const float*)d_in[2];   // [N,H]
    const float* lng  = (const float*)d_in[3];   // [N,H]
    const float* lnb  = (const float*)d_in[4];   // [N,H]
    const float* W2   = (const float*)d_in[5];   // [N,H,D]
    const float* b2   = (const float*)d_in[6];   // [N,D]
    const float* encW = (const float*)d_in[7];   // [D,D]
    const float* encb = (const float*)d_in[8];   // [D]
    const float* decW = (const float*)d_in[9];   // [D,D]
    const float* decb = (const float*)d_in[10];  // [D]
    const float* conn = (const float*)d_in[11];  // [DEPTH,N,N]

    float* xbuf = (float*)d_out;  // x ping buffer; final depth leaves result here
    float* mbuf = (float*)d_ws;   // [B,N,D] message scratch (64 MB)

    const dim3 blk(32);
    const dim3 gA(FN, FB / 16);
    const dim3 gB(FB);

    for (int d = 0; d < 3; ++d) {
        const float* xin = (d == 0) ? x0 : xbuf;
        fractal_stageA<<<gA, blk, 0, stream>>>(xin, W1, b1, lng, lnb, W2, b2,
                                               encW, encb, mbuf);
        fractal_stageB<<<gB, blk, 0, stream>>>(mbuf, conn + (size_t)d * FN * FN,
                                               decW, decb, xbuf);
    }
}